// BarrelShifterRight8_23407571764122
// MI455X (gfx1250) — compile-verified
//
#include <hip/hip_runtime.h>
#include <stdint.h>

#define THREADS 256

// Cooperative block kernel: block b owns rows [b*256, b*256+256).
// X chunk   : 256 rows * 32 B = 8192 B contiguous  -> staged via async B128 copies
// shift chunk: 256 rows * 16 B = 4096 B contiguous -> staged via async B128 copies
__global__ __launch_bounds__(THREADS) void barrel8_main_kernel(
    const float* __restrict__ X, const float* __restrict__ S,
    float* __restrict__ out)
{
    __shared__ __align__(16) float xs[THREADS * 8];  // 8 KB
    __shared__ __align__(16) float ss[THREADS * 4];  // 4 KB

    const int t = threadIdx.x;
    const long long block_row0 = (long long)blockIdx.x * THREADS;

    // ---- Stage global -> LDS with CDNA5 async copies (ASYNCcnt-tracked) ----
    // Per-lane global source addresses (contiguous 16B pieces across the wave).
    const uint8_t* gx = (const uint8_t*)X + block_row0 * 32 + (size_t)t * 16;
    const uint8_t* gs = (const uint8_t*)S + block_row0 * 16 + (size_t)t * 16;
    uint64_t gxa = (uint64_t)(uintptr_t)gx;
    uint64_t gsa = (uint64_t)(uintptr_t)gs;

    // LDS byte offsets: low 32 bits of the generic (flat) address of a __shared__
    // object are the LDS byte address (shared aperture layout).
    uint32_t lx = (uint32_t)(uintptr_t)(void*)(&xs[0]) + (uint32_t)t * 16u;
    uint32_t ls = (uint32_t)(uintptr_t)(void*)(&ss[0]) + (uint32_t)t * 16u;

    // Two B128 pieces of the X chunk (offset:4096 advances BOTH the LDS and the
    // global address per the ISA async-load pseudocode), one piece of shift.
    asm volatile(
        "global_load_async_to_lds_b128 %0, %2, off\n\t"
        "global_load_async_to_lds_b128 %0, %2, off offset:4096\n\t"
        "global_load_async_to_lds_b128 %1, %3, off\n\t"
        "s_wait_asynccnt 0"
        :
        : "v"(lx), "v"(ls), "v"(gxa), "v"(gsa)
        : "memory");
    __syncthreads();  // other waves' async copies must also be visible

    // ---- Per-lane compute: row t of this block's chunk ----
    const float4* xv = (const float4*)xs;
    const float4* sv = (const float4*)ss;
    float4 a = xv[2 * t];      // x0..x3
    float4 b = xv[2 * t + 1];  // x4..x7
    float4 s = sv[t];          // s.x=shift[0](zero) s.y=shift[1](>>4) s.z=shift[2](>>2) s.w=shift[3](>>1)

    float x0 = a.x, x1 = a.y, x2 = a.z, x3 = a.w;
    float x4 = b.x, x5 = b.y, x6 = b.z, x7 = b.w;

    {   // stage 1: shift right by 1 if shift[...,3]
        bool c = s.w > 0.5f;
        x7 = c ? x6 : x7; x6 = c ? x5 : x6; x5 = c ? x4 : x5; x4 = c ? x3 : x4;
        x3 = c ? x2 : x3; x2 = c ? x1 : x2; x1 = c ? x0 : x1; x0 = c ? 0.f : x0;
    }
    {   // stage 2: shift right by 2 if shift[...,2]
        bool c = s.z > 0.5f;
        x7 = c ? x5 : x7; x6 = c ? x4 : x6; x5 = c ? x3 : x5; x4 = c ? x2 : x4;
        x3 = c ? x1 : x3; x2 = c ? x0 : x2; x1 = c ? 0.f : x1; x0 = c ? 0.f : x0;
    }
    {   // stage 3: shift right by 4 if shift[...,1]
        bool c = s.y > 0.5f;
        x7 = c ? x3 : x7; x6 = c ? x2 : x6; x5 = c ? x1 : x5; x4 = c ? x0 : x4;
        x3 = c ? 0.f : x3; x2 = c ? 0.f : x2; x1 = c ? 0.f : x1; x0 = c ? 0.f : x0;
    }
    {   // stage 4: zero everything if shift[...,0]
        bool c = s.x > 0.5f;
        x7 = c ? 0.f : x7; x6 = c ? 0.f : x6; x5 = c ? 0.f : x5; x4 = c ? 0.f : x4;
        x3 = c ? 0.f : x3; x2 = c ? 0.f : x2; x1 = c ? 0.f : x1; x0 = c ? 0.f : x0;
    }

    float4 o0 = make_float4(x0, x1, x2, x3);
    float4 o1 = make_float4(x4, x5, x6, x7);
    float4* ov = (float4*)(out + (block_row0 + t) * 8);
    ov[0] = o0;
    ov[1] = o1;
}

// Tail kernel for rows not covered by full cooperative blocks (none for N=4M,
// but keeps the launcher generic). Direct guarded loads, same math.
__global__ __launch_bounds__(THREADS) void barrel8_tail_kernel(
    const float* __restrict__ X, const float* __restrict__ S,
    float* __restrict__ out, int row0, int n_rows)
{
    int row = row0 + blockIdx.x * THREADS + threadIdx.x;
    if (row >= n_rows) return;

    const float* xr = X + (long long)row * 8;
    const float* sr = S + (long long)row * 4;
    float x0 = xr[0], x1 = xr[1], x2 = xr[2], x3 = xr[3];
    float x4 = xr[4], x5 = xr[5], x6 = xr[6], x7 = xr[7];
    float sz = sr[0], s4 = sr[1], s2 = sr[2], s1 = sr[3];

    { bool c = s1 > 0.5f;
      x7 = c ? x6 : x7; x6 = c ? x5 : x6; x5 = c ? x4 : x5; x4 = c ? x3 : x4;
      x3 = c ? x2 : x3; x2 = c ? x1 : x2; x1 = c ? x0 : x1; x0 = c ? 0.f : x0; }
    { bool c = s2 > 0.5f;
      x7 = c ? x5 : x7; x6 = c ? x4 : x6; x5 = c ? x3 : x5; x4 = c ? x2 : x4;
      x3 = c ? x1 : x3; x2 = c ? x0 : x2; x1 = c ? 0.f : x1; x0 = c ? 0.f : x0; }
    { bool c = s4 > 0.5f;
      x7 = c ? x3 : x7; x6 = c ? x2 : x6; x5 = c ? x1 : x5; x4 = c ? x0 : x4;
      x3 = c ? 0.f : x3; x2 = c ? 0.f : x2; x1 = c ? 0.f : x1; x0 = c ? 0.f : x0; }
    { bool c = sz > 0.5f;
      x7 = c ? 0.f : x7; x6 = c ? 0.f : x6; x5 = c ? 0.f : x5; x4 = c ? 0.f : x4;
      x3 = c ? 0.f : x3; x2 = c ? 0.f : x2; x1 = c ? 0.f : x1; x0 = c ? 0.f : x0; }

    float* o = out + (long long)row * 8;
    o[0] = x0; o[1] = x1; o[2] = x2; o[3] = x3;
    o[4] = x4; o[5] = x5; o[6] = x6; o[7] = x7;
}

extern "C" void kernel_launch(void* const* d_in, const int* in_sizes, int n_in,
                              void* d_out, int out_size, void* d_ws, size_t ws_size,
                              hipStream_t stream) {
    (void)n_in; (void)out_size; (void)d_ws; (void)ws_size;
    const float* X = (const float*)d_in[0];
    const float* S = (const float*)d_in[1];
    float* out = (float*)d_out;

    const int n_rows = in_sizes[0] / 8;       // X is (N, 8)
    const int full_blocks = n_rows / THREADS; // cooperative blocks (256 rows each)
    const int rem = n_rows - full_blocks * THREADS;

    if (full_blocks > 0) {
        barrel8_main_kernel<<<full_blocks, THREADS, 0, stream>>>(X, S, out);
    }
    if (rem > 0) {
        barrel8_tail_kernel<<<1, THREADS, 0, stream>>>(X, S, out,
                                                       full_blocks * THREADS, n_rows);
    }
}